// DeepseekV2MoE_9053791060139
// MI455X (gfx1250) — compile-verified
//
#include <hip/hip_runtime.h>

// ============================================================================
// DeepseekV2 MoE for gfx1250 (MI455X), fp32 WMMA (v_wmma_f32_16x16x4_f32).
//
// Memory-bound problem (~490MB mandatory reads => ~21us @ 23.3TB/s), so we
// keep fp32 end-to-end and use the fp32 WMMA path. Routed experts are
// token-gathered (206 GFLOP instead of the reference's dense 1.65 TFLOP).
// All combination is deterministic (prefix-sum slot lists, fixed-order adds).
//
// v2: K-pair packed LDS tiles (v2f) so every WMMA fragment is one
// ds_load_b64 into an even-aligned VGPR pair -- removes the v_dual_mov
// shuffling seen in the previous round's inner loop.
// ============================================================================

typedef float v2f __attribute__((ext_vector_type(2)));
typedef float v8f __attribute__((ext_vector_type(8)));

#define T_TOK   4096
#define D_HID   2048
#define N_EXP   16
#define I_INT   1024
#define SH_I    2048
#define RSCALE  2.5f
#define GKC     32      // K-chunk staged in LDS (16 k-pairs)

#define XP_S    36      // v2f stride of packed X tile  (32 rows + pad; the tile stores 64 rows)
#define WP_S    132     // v2f stride of packed W slab  (128 cols + pad)

static __device__ __forceinline__ v8f wmma4(v2f a, v2f b, v8f c) {
  // D = A(16x4,f32) * B(4x16,f32) + C(16x16,f32)
  return __builtin_amdgcn_wmma_f32_16x16x4_f32(false, a, false, b, (short)0, c,
                                               false, false);
}

static __device__ __forceinline__ v2f mk2(float a, float b) {
  v2f r; r.x = a; r.y = b; return r;
}

// interleave rows k (ra) and k+1 (rb) for 4 consecutive columns into packed LDS
static __device__ __forceinline__ void store_pair4(v2f* dst, float4 ra, float4 rb) {
  dst[0] = mk2(ra.x, rb.x);
  dst[1] = mk2(ra.y, rb.y);
  dst[2] = mk2(ra.z, rb.z);
  dst[3] = mk2(ra.w, rb.w);
}

// ---------------------------------------------------------------------------
// Kernel 1: router. One wave (32 lanes) per token: 16 logits, sigmoid,
// grouped top-2-of-4-groups, top-4 experts, normalized weights.
// ---------------------------------------------------------------------------
__global__ __launch_bounds__(256) void moe_router_kernel(
    const float* __restrict__ x, const float* __restrict__ gate_w,
    const float* __restrict__ e_bias, int* __restrict__ topk_idx,
    float* __restrict__ topk_w)
{
  const int lane = threadIdx.x & 31;
  const int wave = threadIdx.x >> 5;
  const int t = blockIdx.x * 8 + wave;

  float acc[16];
#pragma unroll
  for (int e = 0; e < 16; ++e) acc[e] = 0.0f;

  const float* xr = x + (size_t)t * D_HID;
  for (int d = lane; d < D_HID; d += 32) {
    const float xv = xr[d];
    const float* gw = gate_w + d * N_EXP;
#pragma unroll
    for (int e = 0; e < 16; ++e) acc[e] = fmaf(xv, gw[e], acc[e]);
  }
#pragma unroll
  for (int off = 16; off > 0; off >>= 1) {
#pragma unroll
    for (int e = 0; e < 16; ++e) acc[e] += __shfl_xor(acc[e], off, 32);
  }

  if (lane == 0) {
    float s[16], sb[16];
#pragma unroll
    for (int e = 0; e < 16; ++e) {
      s[e]  = 1.0f / (1.0f + __expf(-acc[e]));
      sb[e] = s[e] + e_bias[e];
    }
    // group scores: sum of top-2 biased scores within each group of 4
    float gs[4];
#pragma unroll
    for (int g = 0; g < 4; ++g) {
      float m1 = -1e30f, m2 = -1e30f;
#pragma unroll
      for (int j = 0; j < 4; ++j) {
        const float v = sb[g * 4 + j];
        if (v > m1) { m2 = m1; m1 = v; }
        else if (v > m2) { m2 = v; }
      }
      gs[g] = m1 + m2;
    }
    // top-2 groups (strict > => lowest index on ties, like jax top_k)
    int g1 = 0;
    for (int g = 1; g < 4; ++g) if (gs[g] > gs[g1]) g1 = g;
    int g2 = -1;
    for (int g = 0; g < 4; ++g) {
      if (g == g1) continue;
      if (g2 < 0 || gs[g] > gs[g2]) g2 = g;
    }
    // top-4 experts among allowed groups
    unsigned taken = 0;
    int idx[4]; float wv[4]; float wsum = 0.0f;
    for (int k = 0; k < 4; ++k) {
      int bi = 0; float bv = -1e30f; bool found = false;
      for (int e = 0; e < 16; ++e) {
        const int g = e >> 2;
        if (g != g1 && g != g2) continue;
        if (taken & (1u << e)) continue;
        if (!found || sb[e] > bv) { bv = sb[e]; bi = e; found = true; }
      }
      taken |= (1u << bi);
      idx[k] = bi; wv[k] = s[bi]; wsum += s[bi];
    }
    const float inv = 1.0f / wsum;
#pragma unroll
    for (int k = 0; k < 4; ++k) {
      topk_idx[t * 4 + k] = idx[k];
      topk_w[t * 4 + k]   = wv[k] * inv;
    }
  }
}

// ---------------------------------------------------------------------------
// Kernel 2: deterministic per-expert slot lists via LDS prefix sums.
// Single block of 256 threads; each thread owns 16 tokens. Lists come out
// sorted by token id => bitwise-deterministic downstream.
// ---------------------------------------------------------------------------
__global__ __launch_bounds__(256) void moe_build_lists_kernel(
    const int* __restrict__ topk_idx, const float* __restrict__ topk_w,
    int* __restrict__ counts, int* __restrict__ bases,
    int* __restrict__ slot2tok, float* __restrict__ slotW,
    int* __restrict__ tok2slot)
{
  __shared__ int cnt[N_EXP * 256];
  __shared__ int totS[N_EXP];
  __shared__ int baseS[N_EXP];
  const int tid = threadIdx.x;
#pragma unroll
  for (int e = 0; e < N_EXP; ++e) cnt[e * 256 + tid] = 0;
  __syncthreads();

  const int t0 = tid * 16;
  for (int i = 0; i < 16; ++i) {
    const int t = t0 + i;
#pragma unroll
    for (int k = 0; k < 4; ++k) cnt[topk_idx[t * 4 + k] * 256 + tid] += 1;
  }
  __syncthreads();

  if (tid < N_EXP) {             // exclusive prefix over the 256 chunks
    int run = 0;
    for (int j = 0; j < 256; ++j) {
      const int v = cnt[tid * 256 + j];
      cnt[tid * 256 + j] = run;
      run += v;
    }
    totS[tid]  = run;
    counts[tid] = run;
  }
  __syncthreads();
  if (tid == 0) {                // exclusive scan over experts => slot bases
    int b = 0;
    for (int e = 0; e < N_EXP; ++e) { baseS[e] = b; bases[e] = b; b += totS[e]; }
  }
  __syncthreads();

  for (int i = 0; i < 16; ++i) {
    const int t = t0 + i;
#pragma unroll
    for (int k = 0; k < 4; ++k) {
      const int e = topk_idx[t * 4 + k];
      const int p = baseS[e] + cnt[e * 256 + tid];
      cnt[e * 256 + tid] += 1;
      slot2tok[p] = t;
      slotW[p]    = topk_w[t * 4 + k] * RSCALE;  // fold routed_scale here
      tok2slot[t * 4 + k] = p;
    }
  }
}

// ---------------------------------------------------------------------------
// Kernel 3: GEMM1  A = silu(Xg @ W1_gate) * (Xg @ W1_up)
// blockIdx.z: virtual expert (0..15 routed, 16 = shared expert).
// Block tile 64(M) x 128(N), Kc=32; 8 waves as 2x4; each wave accumulates
// 2x2 gate + 2x2 up tiles of v_wmma_f32_16x16x4_f32. LDS tiles are K-pair
// packed v2f so each fragment is a single ds_load_b64.
// ---------------------------------------------------------------------------
__global__ __launch_bounds__(256) void moe_gemm1_kernel(
    const float* __restrict__ x, const float* __restrict__ w_gate_up,
    const float* __restrict__ ws_gate_up, const int* __restrict__ counts,
    const int* __restrict__ bases, const int* __restrict__ slot2tok,
    float* __restrict__ A_r, float* __restrict__ A_s)
{
  __shared__ v2f Xp[(GKC / 2) * XP_S];    // 16 kpairs x 36 (64 rows are stored)
  __shared__ v2f Wgp[(GKC / 2) * WP_S];   // 16 kpairs x 128 cols (+pad)
  __shared__ v2f Wup[(GKC / 2) * WP_S];

  const int e  = blockIdx.z;
  const int mt = blockIdx.x;
  const int nt = blockIdx.y;

  int M, ldw, upOff, Iexp, base;
  const float* W1;
  if (e < N_EXP) {
    M = counts[e]; base = bases[e];
    ldw = 2 * I_INT; upOff = I_INT; Iexp = I_INT;
    W1 = w_gate_up + (size_t)e * D_HID * (2 * I_INT);
  } else {
    M = T_TOK; base = 0; ldw = 2 * SH_I; upOff = SH_I; Iexp = SH_I;
    W1 = ws_gate_up;
  }
  if (mt * 64 >= M || nt * 128 >= Iexp) return;   // uniform per block

  const int tid  = threadIdx.x;
  const int lane = tid & 31;
  const int wave = tid >> 5;
  const int wm = (wave >> 2) * 32;                // 2 waves along M
  const int wn = (wave & 3) * 32;                 // 4 waves along N

  // ---- X-tile loader mapping (rows lm / lm+32, 4 consecutive k) ----
  const int lk4 = (tid & 7) * 4;                  // k offset within chunk
  const int kp0 = (tid & 7) * 2;                  // kpair index
  const int lm  = tid >> 3;                       // 0..31
  int tok0, tok1;
  {
    const int r0 = mt * 64 + lm;
    const int r1 = r0 + 32;
    if (e < N_EXP) {
      tok0 = (r0 < M) ? slot2tok[base + r0] : 0;  // junk rows masked on store
      tok1 = (r1 < M) ? slot2tok[base + r1] : 0;
    } else { tok0 = r0; tok1 = r1; }
  }
  const float* xr0 = x + (size_t)tok0 * D_HID;
  const float* xr1 = x + (size_t)tok1 * D_HID;

  // ---- W-slab loader mapping: rows k,k+1 for 4 consecutive cols ----
  const int c4 = (tid & 31) * 4;                  // col within 128-tile
  const int kq = tid >> 5;                        // kpair 0..7 (and kq+8)
  const float* Wbase = W1 + (size_t)nt * 128;

  v8f G[2][2] = {};
  v8f U[2][2] = {};

  const int nChunks = D_HID / GKC;                // 64
  for (int ch = 0; ch < nChunks; ++ch) {
    const int k0 = ch * GKC;
    const float4 xv0 = *(const float4*)(xr0 + k0 + lk4);
    const float4 xv1 = *(const float4*)(xr1 + k0 + lk4);
    const float* wr0 = Wbase + (size_t)(k0 + 2 * kq) * ldw + c4;
    const float* wr1 = Wbase + (size_t)(k0 + 2 * (kq + 8)) * ldw + c4;
    const float4 g0a = *(const float4*)(wr0);
    const float4 g0b = *(const float4*)(wr0 + ldw);
    const float4 u0a = *(const float4*)(wr0 + upOff);
    const float4 u0b = *(const float4*)(wr0 + ldw + upOff);
    const float4 g1a = *(const float4*)(wr1);
    const float4 g1b = *(const float4*)(wr1 + ldw);
    const float4 u1a = *(const float4*)(wr1 + upOff);
    const float4 u1b = *(const float4*)(wr1 + ldw + upOff);
    if (ch + 1 < nChunks) {                       // stream-ahead hint
      __builtin_prefetch(wr0 + (size_t)GKC * ldw, 0, 0);
      __builtin_prefetch(xr0 + k0 + GKC + lk4, 0, 0);
    }
    __syncthreads();                              // WAR vs previous compute
    // packed X stores: {x[k], x[k+1]} per row
    Xp[(kp0 + 0) * XP_S + lm]      = mk2(xv0.x, xv0.y);
    Xp[(kp0 + 1) * XP_S + lm]      = mk2(xv0.z, xv0.w);
    Xp[(kp0 + 0) * XP_S + lm + 32] = mk2(xv1.x, xv1.y);
    Xp[(kp0 + 1) * XP_S + lm + 32] = mk2(xv1.z, xv1.w);
    // packed W stores: interleave rows k,k+1
    store_pair4(&Wgp[kq * WP_S + c4],       g0a, g0b);
    store_pair4(&Wgp[(kq + 8) * WP_S + c4], g1a, g1b);
    store_pair4(&Wup[kq * WP_S + c4],       u0a, u0b);
    store_pair4(&Wup[(kq + 8) * WP_S + c4], u1a, u1b);
    __syncthreads();

    const int hk = lane >> 4;                     // kpair offset 0/1 per half
    const int ml = wm + (lane & 15);
    const int nl = wn + (lane & 15);
#pragma unroll
    for (int ks = 0; ks < GKC / 4; ++ks) {
      const int kp = ks * 2 + hk;
      const v2f a0  = Xp[kp * XP_S + ml];
      const v2f a1  = Xp[kp * XP_S + ml + 16];
      const v2f bg0 = Wgp[kp * WP_S + nl];
      const v2f bg1 = Wgp[kp * WP_S + nl + 16];
      const v2f bu0 = Wup[kp * WP_S + nl];
      const v2f bu1 = Wup[kp * WP_S + nl + 16];
      G[0][0] = wmma4(a0, bg0, G[0][0]);
      G[0][1] = wmma4(a0, bg1, G[0][1]);
      G[1][0] = wmma4(a1, bg0, G[1][0]);
      G[1][1] = wmma4(a1, bg1, G[1][1]);
      U[0][0] = wmma4(a0, bu0, U[0][0]);
      U[0][1] = wmma4(a0, bu1, U[0][1]);
      U[1][0] = wmma4(a1, bu0, U[1][0]);
      U[1][1] = wmma4(a1, bu1, U[1][1]);
    }
  }

  // epilogue: silu(g)*u, scatter to A scratch (row-masked by count)
  const int h = lane >> 4;
  const int nlan = lane & 15;
#pragma unroll
  for (int im = 0; im < 2; ++im) {
#pragma unroll
    for (int in = 0; in < 2; ++in) {
      const int gcol = nt * 128 + wn + in * 16 + nlan;
#pragma unroll
      for (int v = 0; v < 8; ++v) {
        const int rowe = mt * 64 + wm + im * 16 + v + 8 * h;
        if (rowe < M) {
          const float g = G[im][in][v];
          const float u = U[im][in][v];
          const float a = (g / (1.0f + __expf(-g))) * u;
          if (e < N_EXP) A_r[(size_t)(base + rowe) * I_INT + gcol] = a;
          else           A_s[(size_t)rowe * SH_I + gcol] = a;
        }
      }
    }
  }
}

// ---------------------------------------------------------------------------
// Kernel 4: GEMM2  Y = (A @ W2) * rowWeight   (routed -> Y scratch)
//                  out = A_s @ ws_down        (shared -> d_out directly)
// ---------------------------------------------------------------------------
__global__ __launch_bounds__(256) void moe_gemm2_kernel(
    const float* __restrict__ A_r, const float* __restrict__ A_s,
    const float* __restrict__ w_down, const float* __restrict__ ws_down,
    const int* __restrict__ counts, const int* __restrict__ bases,
    const float* __restrict__ slotW, float* __restrict__ Y,
    float* __restrict__ outp)
{
  __shared__ v2f Ap[(GKC / 2) * XP_S];
  __shared__ v2f Wp[(GKC / 2) * WP_S];

  const int e  = blockIdx.z;
  const int mt = blockIdx.x;
  const int nt = blockIdx.y;

  int M, K, base;
  const float* Aexp; const float* W2;
  if (e < N_EXP) {
    M = counts[e]; base = bases[e]; K = I_INT;
    Aexp = A_r + (size_t)base * I_INT;
    W2 = w_down + (size_t)e * I_INT * D_HID;
  } else {
    M = T_TOK; base = 0; K = SH_I; Aexp = A_s; W2 = ws_down;
  }
  if (mt * 64 >= M) return;                       // nt always valid (D/128=16)

  const int tid  = threadIdx.x;
  const int lane = tid & 31;
  const int wave = tid >> 5;
  const int wm = (wave >> 2) * 32;
  const int wn = (wave & 3) * 32;

  const int lk4 = (tid & 7) * 4;
  const int kp0 = (tid & 7) * 2;
  const int lm  = tid >> 3;
  int r0 = mt * 64 + lm;       if (r0 >= M) r0 = M - 1;   // clamp loads
  int r1 = mt * 64 + lm + 32;  if (r1 >= M) r1 = M - 1;
  const float* ar0 = Aexp + (size_t)r0 * K;
  const float* ar1 = Aexp + (size_t)r1 * K;

  const int c4 = (tid & 31) * 4;
  const int kq = tid >> 5;
  const float* Wbase = W2 + (size_t)nt * 128;

  v8f C[2][2] = {};
  const int nChunks = K / GKC;
  for (int ch = 0; ch < nChunks; ++ch) {
    const int k0 = ch * GKC;
    const float4 av0 = *(const float4*)(ar0 + k0 + lk4);
    const float4 av1 = *(const float4*)(ar1 + k0 + lk4);
    const float* wr0 = Wbase + (size_t)(k0 + 2 * kq) * D_HID + c4;
    const float* wr1 = Wbase + (size_t)(k0 + 2 * (kq + 8)) * D_HID + c4;
    const float4 w0a = *(const float4*)(wr0);
    const float4 w0b = *(const float4*)(wr0 + D_HID);
    const float4 w1a = *(const float4*)(wr1);
    const float4 w1b = *(const float4*)(wr1 + D_HID);
    if (ch + 1 < nChunks)
      __builtin_prefetch(wr0 + (size_t)GKC * D_HID, 0, 0);

    __syncthreads();
    Ap[(kp0 + 0) * XP_S + lm]      = mk2(av0.x, av0.y);
    Ap[(kp0 + 1) * XP_S + lm]      = mk2(av0.z, av0.w);
    Ap[(kp0 + 0) * XP_S + lm + 32] = mk2(av1.x, av1.y);
    Ap[(kp0 + 1) * XP_S + lm + 32] = mk2(av1.z, av1.w);
    store_pair4(&Wp[kq * WP_S + c4],       w0a, w0b);
    store_pair4(&Wp[(kq + 8) * WP_S + c4], w1a, w1b);
    __syncthreads();

    const int hk = lane >> 4;
    const int ml = wm + (lane & 15);
    const int nl = wn + (lane & 15);
#pragma unroll
    for (int ks = 0; ks < GKC / 4; ++ks) {
      const int kp = ks * 2 + hk;
      const v2f a0 = Ap[kp * XP_S + ml];
      const v2f a1 = Ap[kp * XP_S + ml + 16];
      const v2f b0 = Wp[kp * WP_S + nl];
      const v2f b1 = Wp[kp * WP_S + nl + 16];
      C[0][0] = wmma4(a0, b0, C[0][0]);
      C[0][1] = wmma4(a0, b1, C[0][1]);
      C[1][0] = wmma4(a1, b0, C[1][0]);
      C[1][1] = wmma4(a1, b1, C[1][1]);
    }
  }

  const int h = lane >> 4;
  const int nlan = lane & 15;
#pragma unroll
  for (int im = 0; im < 2; ++im) {
#pragma unroll
    for (int in = 0; in < 2; ++in) {
      const int gcol = nt * 128 + wn + in * 16 + nlan;
#pragma unroll
      for (int v = 0; v < 8; ++v) {
        const int rowe = mt * 64 + wm + im * 16 + v + 8 * h;
        if (rowe < M) {
          if (e < N_EXP) {
            const float w = slotW[base + rowe];
            Y[(size_t)(base + rowe) * D_HID + gcol] = C[im][in][v] * w;
          } else {
            outp[(size_t)rowe * D_HID + gcol] = C[im][in][v];
          }
        }
      }
    }
  }
}

// ---------------------------------------------------------------------------
// Kernel 5: combine. out[t] (already = shared) += sum_k Y[slot_k(t)], in
// fixed k order => deterministic fp32 result.
// ---------------------------------------------------------------------------
__global__ __launch_bounds__(256) void moe_combine_kernel(
    const float* __restrict__ Y, const int* __restrict__ tok2slot,
    float* __restrict__ outp)
{
  const int t = blockIdx.x;
  const int c = threadIdx.x * 8;
  float* orow = outp + (size_t)t * D_HID + c;
  float4 o0 = *(float4*)(orow);
  float4 o1 = *(float4*)(orow + 4);
#pragma unroll
  for (int k = 0; k < 4; ++k) {
    const int s = tok2slot[t * 4 + k];
    const float* yr = Y + (size_t)s * D_HID + c;
    const float4 y0 = *(const float4*)(yr);
    const float4 y1 = *(const float4*)(yr + 4);
    o0.x += y0.x; o0.y += y0.y; o0.z += y0.z; o0.w += y0.w;
    o1.x += y1.x; o1.y += y1.y; o1.z += y1.z; o1.w += y1.w;
  }
  *(float4*)(orow)     = o0;
  *(float4*)(orow + 4) = o1;
}

// ---------------------------------------------------------------------------
extern "C" void kernel_launch(void* const* d_in, const int* in_sizes, int n_in,
                              void* d_out, int out_size, void* d_ws, size_t ws_size,
                              hipStream_t stream) {
  (void)in_sizes; (void)n_in; (void)out_size; (void)ws_size;
  const float* x          = (const float*)d_in[0];
  const float* gate_w     = (const float*)d_in[1];
  const float* e_bias     = (const float*)d_in[2];
  const float* w_gate_up  = (const float*)d_in[3];
  const float* w_down     = (const float*)d_in[4];
  const float* ws_gate_up = (const float*)d_in[5];
  const float* ws_down    = (const float*)d_in[6];
  float* outp = (float*)d_out;

  // workspace carve-up (deterministic layout)
  char* p = (char*)d_ws;
  auto alloc = [&](size_t bytes) {
    void* r = (void*)p;
    p += (bytes + 255) & ~(size_t)255;
    return r;
  };
  int*   topk_idx = (int*)  alloc((size_t)T_TOK * 4 * sizeof(int));
  float* topk_w   = (float*)alloc((size_t)T_TOK * 4 * sizeof(float));
  int*   counts   = (int*)  alloc(N_EXP * sizeof(int));
  int*   bases    = (int*)  alloc(N_EXP * sizeof(int));
  int*   slot2tok = (int*)  alloc((size_t)T_TOK * 4 * sizeof(int));
  float* slotW    = (float*)alloc((size_t)T_TOK * 4 * sizeof(float));
  int*   tok2slot = (int*)  alloc((size_t)T_TOK * 4 * sizeof(int));
  float* A_r      = (float*)alloc((size_t)T_TOK * 4 * I_INT * sizeof(float));
  float* A_s      = (float*)alloc((size_t)T_TOK * SH_I * sizeof(float));
  float* Y        = (float*)alloc((size_t)T_TOK * 4 * D_HID * sizeof(float));

  moe_router_kernel<<<dim3(T_TOK / 8), dim3(256), 0, stream>>>(
      x, gate_w, e_bias, topk_idx, topk_w);

  moe_build_lists_kernel<<<dim3(1), dim3(256), 0, stream>>>(
      topk_idx, topk_w, counts, bases, slot2tok, slotW, tok2slot);

  moe_gemm1_kernel<<<dim3(T_TOK / 64, 16, N_EXP + 1), dim3(256), 0, stream>>>(
      x, w_gate_up, ws_gate_up, counts, bases, slot2tok, A_r, A_s);

  moe_gemm2_kernel<<<dim3(T_TOK / 64, D_HID / 128, N_EXP + 1), dim3(256), 0, stream>>>(
      A_r, A_s, w_down, ws_down, counts, bases, slotW, Y, outp);

  moe_combine_kernel<<<dim3(T_TOK), dim3(256), 0, stream>>>(Y, tok2slot, outp);
}